// DecoderRNN_73615739453618
// MI455X (gfx1250) — compile-verified
//
#include <hip/hip_runtime.h>
#include <math.h>

// Show-Attend-Tell decoder for MI455X (gfx1250, wave32).
// fp32 WMMA (V_WMMA_F32_16X16X4_F32) for all GEMMs; weights are L2-resident
// (105MB < 192MB L2), sequence is latency-bound -> stream-ordered kernel chain.
// Each wave computes a 64x16 output slab (4 M-tiles x 1 N-tile), reusing the
// B (weight) fragment 4x per K-step; the 4 WMMAs target independent
// accumulators so they pipeline without WMMA->WMMA RAW hazards.

typedef float v2f __attribute__((ext_vector_type(2)));
typedef float v8f __attribute__((ext_vector_type(8)));

#define BB    64
#define PP    49
#define ENC   2048
#define DEC   512
#define ATT   512
#define EMB   512
#define VOC   10000
#define LL    33
#define TT    32   // L-1 steps

// ---------------------------------------------------------------------------
// NT GEMM via f32 WMMA: C[m,n] = sum_k A[m,k]*W[n,k] + bias[n] (+C if acc)
// A: M x K (row-major, lda), W: N x K (row-major, ldw), C row stride ldc.
// Requires M % 64 == 0, N % 16 == 0, K % 4 == 0 (true for all call sites).
// Grid: (M/64, ceil(N/64)); block = 128 (4 waves, one N-tile each).
// ISA layouts (CDNA5 ISA):
//   A 16x4 f32: lane l holds row m+(l&15), K = 2*(l>>4)+{0,1}  (2 VGPRs)
//   B 4x16 f32: lane l holds col n+(l&15), K = 2*(l>>4)+{0,1}  (2 VGPRs)
//   C/D 16x16:  vgpr v, lane l -> row m+v+8*(l>>4), col n+(l&15)
// ---------------------------------------------------------------------------
__global__ __launch_bounds__(128) void gemm_f32_wmma(
    const float* __restrict__ A, int lda,
    const float* __restrict__ W, int ldw,
    const float* __restrict__ bias,
    float* __restrict__ C, long ldc,
    int N, int K, int acc)
{
    const int lane  = threadIdx.x & 31;
    const int wave  = threadIdx.x >> 5;
    const int tileN = blockIdx.y * 4 + wave;
    if (tileN * 16 >= N) return;

    const int m0 = blockIdx.x * 64;   // 4 consecutive M-tiles per wave
    const int n0 = tileN * 16;
    const int r  = lane & 15;
    const int hf = lane >> 4;

    const float* arow = A + (size_t)(m0 + r) * lda + 2 * hf;
    const float* wrow = W + (size_t)(n0 + r) * ldw + 2 * hf;
    const size_t astep = (size_t)16 * lda;     // next M tile

    v8f c0, c1, c2, c3;
    const float bv = bias[n0 + r];
    if (acc) {
        #pragma unroll
        for (int v = 0; v < 8; ++v) {
            const size_t row = (size_t)(m0 + v + 8 * hf) * ldc + (n0 + r);
            c0[v] = C[row]                    + bv;
            c1[v] = C[row + 16 * (size_t)ldc] + bv;
            c2[v] = C[row + 32 * (size_t)ldc] + bv;
            c3[v] = C[row + 48 * (size_t)ldc] + bv;
        }
    } else {
        #pragma unroll
        for (int v = 0; v < 8; ++v) { c0[v] = bv; c1[v] = bv; c2[v] = bv; c3[v] = bv; }
    }

    #pragma unroll 4
    for (int k0 = 0; k0 < K; k0 += 4) {
        v2f b  = *(const v2f*)(wrow + k0);                 // shared weight frag
        v2f a0 = *(const v2f*)(arow + k0);
        v2f a1 = *(const v2f*)(arow + astep     + k0);
        v2f a2 = *(const v2f*)(arow + astep * 2 + k0);
        v2f a3 = *(const v2f*)(arow + astep * 3 + k0);
        // (neg_a, A, neg_b, B, c_mod, C, reuse_a, reuse_b)
        c0 = __builtin_amdgcn_wmma_f32_16x16x4_f32(false, a0, false, b, (short)0, c0, false, false);
        c1 = __builtin_amdgcn_wmma_f32_16x16x4_f32(false, a1, false, b, (short)0, c1, false, false);
        c2 = __builtin_amdgcn_wmma_f32_16x16x4_f32(false, a2, false, b, (short)0, c2, false, false);
        c3 = __builtin_amdgcn_wmma_f32_16x16x4_f32(false, a3, false, b, (short)0, c3, false, false);
    }

    #pragma unroll
    for (int v = 0; v < 8; ++v) {
        const size_t row = (size_t)(m0 + v + 8 * hf) * ldc + (n0 + r);
        C[row]                    = c0[v];
        C[row + 16 * (size_t)ldc] = c1[v];
        C[row + 32 * (size_t)ldc] = c2[v];
        C[row + 48 * (size_t)ldc] = c3[v];
    }
}

// ---------------------------------------------------------------------------
__global__ void mean_kernel(const float* __restrict__ features,
                            float* __restrict__ mean_enc)
{
    int idx = blockIdx.x * blockDim.x + threadIdx.x;
    if (idx >= BB * ENC) return;
    int b = idx / ENC, e = idx % ENC;
    float s = 0.f;
    for (int p = 0; p < PP; ++p)
        s += features[((size_t)b * PP + p) * ENC + e];
    mean_enc[idx] = s * (1.0f / PP);
}

__global__ void embed_kernel(const int* __restrict__ captions,
                             const float* __restrict__ emb_W,
                             float* __restrict__ embeds)
{
    int idx = blockIdx.x * blockDim.x + threadIdx.x;
    if (idx >= BB * TT * EMB) return;
    int b   = idx / (TT * EMB);
    int rem = idx % (TT * EMB);
    int t   = rem / EMB;
    int j   = rem % EMB;
    int tok = captions[b * LL + t];   // captions[:, :-1]
    embeds[idx] = emb_W[(size_t)tok * EMB + j];
}

// Attention: scores -> softmax over P -> context; builds lstm_in=[x_t|context],
// writes alpha to output. One block per batch row.
__global__ __launch_bounds__(256) void attn_kernel(
    const float* __restrict__ u_hs,     // B x ATT
    const float* __restrict__ w_ah,     // B x P x ATT
    const float* __restrict__ A_w,      // ATT
    const float* __restrict__ A_b,      // 1
    const float* __restrict__ features, // B x P x ENC
    const float* __restrict__ embeds,   // B x T x EMB
    float* __restrict__ lstm_in,        // B x (EMB+ENC)
    float* __restrict__ alphas_out,     // B x T x P
    int t)
{
    __shared__ float sc[PP];
    const int b    = blockIdx.x;
    const int tid  = threadIdx.x;
    const int lane = tid & 31;
    const int wv   = tid >> 5;          // 8 waves

    const float* uh = u_hs + (size_t)b * ATT;
    for (int p = wv; p < PP; p += 8) {
        const float* wa = w_ah + ((size_t)b * PP + p) * ATT;
        float s = 0.f;
        for (int e = lane; e < ATT; e += 32)
            s += A_w[e] * tanhf(uh[e] + wa[e]);
        #pragma unroll
        for (int off = 16; off > 0; off >>= 1)
            s += __shfl_xor(s, off, 32);
        if (lane == 0) sc[p] = s + A_b[0];
    }
    __syncthreads();

    if (tid == 0) {   // softmax over 49 entries: trivial serial cost
        float mx = -1e30f;
        for (int p = 0; p < PP; ++p) mx = fmaxf(mx, sc[p]);
        float sum = 0.f;
        for (int p = 0; p < PP; ++p) { float e = expf(sc[p] - mx); sc[p] = e; sum += e; }
        float inv = 1.f / sum;
        for (int p = 0; p < PP; ++p) {
            sc[p] *= inv;
            alphas_out[((size_t)b * TT + t) * PP + p] = sc[p];
        }
    }
    __syncthreads();

    // context = alpha @ features  -> lstm_in[:, EMB:]
    for (int e = tid; e < ENC; e += 256) {
        const float* f = features + (size_t)b * PP * ENC + e;
        float acc = 0.f;
        for (int p = 0; p < PP; ++p) acc += sc[p] * f[(size_t)p * ENC];
        lstm_in[(size_t)b * (EMB + ENC) + EMB + e] = acc;
    }
    // x_t -> lstm_in[:, :EMB]
    for (int j = tid; j < EMB; j += 256)
        lstm_in[(size_t)b * (EMB + ENC) + j] = embeds[((size_t)b * TT + t) * EMB + j];
}

__global__ void lstm_cell(const float* __restrict__ gates,
                          float* __restrict__ h, float* __restrict__ c)
{
    int idx = blockIdx.x * blockDim.x + threadIdx.x;
    if (idx >= BB * DEC) return;
    int b = idx / DEC, j = idx % DEC;
    const float* g = gates + (size_t)b * 4 * DEC;
    float i  = 1.f / (1.f + expf(-g[j]));
    float f  = 1.f / (1.f + expf(-g[DEC + j]));
    float gg = tanhf(g[2 * DEC + j]);
    float o  = 1.f / (1.f + expf(-g[3 * DEC + j]));
    float cn = f * c[idx] + i * gg;
    c[idx] = cn;
    h[idx] = o * tanhf(cn);
}

// ---------------------------------------------------------------------------
extern "C" void kernel_launch(void* const* d_in, const int* in_sizes, int n_in,
                              void* d_out, int out_size, void* d_ws, size_t ws_size,
                              hipStream_t stream)
{
    const float* features = (const float*)d_in[0];
    const int*   captions = (const int*)  d_in[1];
    const float* emb_W    = (const float*)d_in[2];
    const float* W_w      = (const float*)d_in[3];
    const float* W_b      = (const float*)d_in[4];
    const float* U_w      = (const float*)d_in[5];
    const float* U_b      = (const float*)d_in[6];
    const float* A_w      = (const float*)d_in[7];
    const float* A_b      = (const float*)d_in[8];
    const float* ih_w     = (const float*)d_in[9];
    const float* ih_b     = (const float*)d_in[10];
    const float* ic_w     = (const float*)d_in[11];
    const float* ic_b     = (const float*)d_in[12];
    const float* W_ih     = (const float*)d_in[13];
    const float* W_hh     = (const float*)d_in[14];
    const float* b_ih     = (const float*)d_in[15];
    const float* b_hh     = (const float*)d_in[16];
    const float* fcn_w    = (const float*)d_in[17];
    const float* fcn_b    = (const float*)d_in[18];

    float* preds  = (float*)d_out;                           // B x T x V
    float* alphas = (float*)d_out + (size_t)BB * TT * VOC;   // B x T x P

    float* ws       = (float*)d_ws;
    float* mean_enc = ws;  ws += (size_t)BB * ENC;
    float* h        = ws;  ws += (size_t)BB * DEC;
    float* c        = ws;  ws += (size_t)BB * DEC;
    float* u_hs     = ws;  ws += (size_t)BB * ATT;
    float* embeds   = ws;  ws += (size_t)BB * TT * EMB;
    float* w_ah     = ws;  ws += (size_t)BB * PP * ATT;
    float* lstm_in  = ws;  ws += (size_t)BB * (EMB + ENC);
    float* gates    = ws;  ws += (size_t)BB * 4 * DEC;

    auto gemm = [&](const float* A, int lda, const float* W, int ldw,
                    const float* bias, float* C, long ldc,
                    int M, int N, int K, int acc) {
        dim3 grid(M / 64, (N / 16 + 3) / 4);
        gemm_f32_wmma<<<grid, 128, 0, stream>>>(A, lda, W, ldw, bias, C, ldc, N, K, acc);
    };

    // ---- precompute ----
    mean_kernel<<<(BB * ENC + 255) / 256, 256, 0, stream>>>(features, mean_enc);
    gemm(mean_enc, ENC, ih_w, ENC, ih_b, h, DEC, BB, DEC, ENC, 0);        // h0
    gemm(mean_enc, ENC, ic_w, ENC, ic_b, c, DEC, BB, DEC, ENC, 0);        // c0
    embed_kernel<<<(BB * TT * EMB + 255) / 256, 256, 0, stream>>>(captions, emb_W, embeds);
    gemm(features, ENC, W_w, ENC, W_b, w_ah, ATT, BB * PP, ATT, ENC, 0);  // w_ah

    // ---- 32-step recurrent scan ----
    for (int t = 0; t < TT; ++t) {
        gemm(h, DEC, U_w, DEC, U_b, u_hs, ATT, BB, ATT, DEC, 0);          // u_hs
        attn_kernel<<<BB, 256, 0, stream>>>(u_hs, w_ah, A_w, A_b, features,
                                            embeds, lstm_in, alphas, t);
        gemm(lstm_in, EMB + ENC, W_ih, EMB + ENC, b_ih,
             gates, 4 * DEC, BB, 4 * DEC, EMB + ENC, 0);                  // lstm_in @ W_ih^T
        gemm(h, DEC, W_hh, DEC, b_hh,
             gates, 4 * DEC, BB, 4 * DEC, DEC, 1);                        // += h @ W_hh^T
        lstm_cell<<<(BB * DEC + 255) / 256, 256, 0, stream>>>(gates, h, c);
        gemm(h, DEC, fcn_w, DEC, fcn_b,
             preds + (size_t)t * VOC, (long)TT * VOC, BB, VOC, DEC, 0);   // logits
    }
}